// SynthesisGroupConv2d_67774583931171
// MI455X (gfx1250) — compile-verified
//
#include <hip/hip_runtime.h>
#include <hip/hip_bf16.h>

typedef float v2f __attribute__((ext_vector_type(2)));
typedef float v4f __attribute__((ext_vector_type(4)));
typedef float v8f __attribute__((ext_vector_type(8)));

#define B_    16
#define CI_   256
#define CO_   256
#define HW_   4096
#define WDIM_ 512
#define NF_   16
#define FD_   512
#define KP_   2304          // CI * 3 * 3
#define LRM   (1.0f/64.0f)
#define KC    32            // K chunk staged in LDS
#define LDA   36            // padded LDS stride: even (b64 frags) + mult-of-4 (b128 stores)

// ---------------- kernel 1: affine  s[b][k] -------------------------------
__global__ __launch_bounds__(256)
void affine_k(const float* __restrict__ temb, const float* __restrict__ aw,
              const float* __restrict__ ab, float* __restrict__ s) {
  const int t = threadIdx.x;            // 256 = 16 b * 16 k
  const int b = t >> 4, k = t & 15;
  const float wg = LRM * 0.04419417382415922f;   // (1/SR)/sqrt(WDIM)
  const float* tr = temb + b * WDIM_;
  const float* wr = aw + k * WDIM_;
  float acc = 0.f;
  for (int w = 0; w < WDIM_; ++w) acc += tr[w] * wr[w];
  s[t] = acc * wg + ab[k] * LRM;
}

// ---------------- kernel 2: mag_b[b][i][f] --------------------------------
__global__ __launch_bounds__(256)
void mag_k(const float* __restrict__ magnitude, const float* __restrict__ s,
           float* __restrict__ mag) {
  const int i = blockIdx.x, b = blockIdx.y;
  __shared__ float sv[NF_];
  if (threadIdx.x < NF_) sv[threadIdx.x] = s[b * NF_ + threadIdx.x];
  __syncthreads();
  for (int f = threadIdx.x; f < FD_; f += 256) {
    float acc = 0.f;
#pragma unroll
    for (int k = 0; k < NF_; ++k)
      acc += magnitude[(size_t)(k * CI_ + i) * FD_ + f] * sv[k];
    mag[(size_t)(b * CI_ + i) * FD_ + f] = acc * 0.25f;   // 1/sqrt(NUM_FREQ)
  }
}

// WMMA helper: D = A(16x4) * B(4x16) + C, f32
__device__ __forceinline__ v8f wmma4(v2f a, v2f b, v8f c) {
  return __builtin_amdgcn_wmma_f32_16x16x4_f32(false, a, false, b, (short)0, c,
                                               false, false);
}

// ---- shared staging helpers (B tile: 64 rows x KC, contiguous k) ---------
__device__ __forceinline__ void loadB_regs(const float* __restrict__ src,
                                           size_t row_stride, int n0, int k,
                                           int t, v4f bR[2]) {
  const int kl4 = (t & 7) * 4, nr = t >> 3;     // 32 rows per pass
#pragma unroll
  for (int i = 0; i < 2; ++i)
    bR[i] = *(const v4f*)(src + (size_t)(n0 + nr + i * 32) * row_stride + k + kl4);
}
__device__ __forceinline__ void storeB_lds(float (*Bs)[LDA], int t, const v4f bR[2]) {
  const int kl4 = (t & 7) * 4, nr = t >> 3;
#pragma unroll
  for (int i = 0; i < 2; ++i)
    *(v4f*)&Bs[nr + i * 32][kl4] = bR[i];
}

// ---------------- kernel 3: per-batch kernel synthesis GEMM ---------------
// C[p,o] = sum_f basis[p,f]*mag[b,p/9,f] * out_linear[o,f], store ker[b][o][p]
__device__ __forceinline__ void gen_loadA(const float* __restrict__ basis,
                                          const float* __restrict__ mag,
                                          int b, int p0, int f, int t, v4f aR[4]) {
  const int fl4 = (t & 7) * 4, mr = t >> 3;     // 32 rows per pass
#pragma unroll
  for (int i = 0; i < 4; ++i) {
    const int p = p0 + mr + i * 32;
    v4f bv = *(const v4f*)(basis + (size_t)p * FD_ + f + fl4);
    v4f mv = *(const v4f*)(mag + (size_t)(b * CI_ + p / 9) * FD_ + f + fl4);
#pragma unroll
    for (int c = 0; c < 4; ++c) bv[c] *= mv[c];
    aR[i] = bv;
  }
}
__device__ __forceinline__ void gen_storeA(float (*As)[LDA], int t, const v4f aR[4]) {
  const int fl4 = (t & 7) * 4, mr = t >> 3;
#pragma unroll
  for (int i = 0; i < 4; ++i)
    *(v4f*)&As[mr + i * 32][fl4] = aR[i];
}

__global__ __launch_bounds__(256)
void gen_k(const float* __restrict__ basis, const float* __restrict__ mag,
           const float* __restrict__ outlin, float* __restrict__ ker) {
  __shared__ float As[128][LDA];
  __shared__ float Bs[64][LDA];
  const int mtile = blockIdx.x, ntile = blockIdx.y, b = blockIdx.z;
  const int t = threadIdx.x, lane = t & 31, wid = t >> 5;
  const int wm = wid & 3, wn = wid >> 2;
  const int half = lane >> 4, lrow = lane & 15;
  const int p0 = mtile * 128, n0 = ntile * 64;

  v8f acc[2][2] = {};
  v4f aR[4], bR[2];
  gen_loadA(basis, mag, b, p0, 0, t, aR);
  loadB_regs(outlin, FD_, n0, 0, t, bR);
  const int NCH = FD_ / KC;
  for (int kc = 0; kc < NCH; ++kc) {
    gen_storeA(As, t, aR);
    storeB_lds(Bs, t, bR);
    __syncthreads();
    if (kc + 1 < NCH) {                 // prefetch next chunk during compute
      gen_loadA(basis, mag, b, p0, (kc + 1) * KC, t, aR);
      loadB_regs(outlin, FD_, n0, (kc + 1) * KC, t, bR);
    }
#pragma unroll
    for (int kk = 0; kk < KC; kk += 4) {
      v2f a0 = *(const v2f*)&As[wm * 32 + lrow][kk + 2 * half];
      v2f a1 = *(const v2f*)&As[wm * 32 + 16 + lrow][kk + 2 * half];
      v2f b0 = *(const v2f*)&Bs[wn * 32 + lrow][kk + 2 * half];
      v2f b1 = *(const v2f*)&Bs[wn * 32 + 16 + lrow][kk + 2 * half];
      acc[0][0] = wmma4(a0, b0, acc[0][0]);
      acc[0][1] = wmma4(a0, b1, acc[0][1]);
      acc[1][0] = wmma4(a1, b0, acc[1][0]);
      acc[1][1] = wmma4(a1, b1, acc[1][1]);
    }
    __syncthreads();
  }
  const float scale = 1.0f / 1536.0f;   // FREQ_GAIN * GAIN
#pragma unroll
  for (int mi = 0; mi < 2; ++mi)
#pragma unroll
    for (int ni = 0; ni < 2; ++ni) {
      const int o = n0 + wn * 32 + ni * 16 + lrow;
      const int p = p0 + wm * 32 + mi * 16 + half * 8;
      float* dst = ker + (size_t)(b * CO_ + o) * KP_ + p;
      v8f v = acc[mi][ni];
      v4f lo, hi;
#pragma unroll
      for (int r = 0; r < 4; ++r) { lo[r] = v[r] * scale; hi[r] = v[r + 4] * scale; }
      *(v4f*)dst = lo;
      *(v4f*)(dst + 4) = hi;
    }
}

// ---------------- kernel 4: implicit-GEMM grouped conv + silu -------------
__device__ __forceinline__ void conv_gather(const float* __restrict__ x, int b,
                                            int m0, int k0, int t, float aR[16]) {
  const int ml0 = t & 31, kr = t >> 5;
#pragma unroll
  for (int ko = 0; ko < 4; ++ko) {
    const int k = k0 + kr + ko * 8;
    const int ich = k / 9;
    const int r = k - ich * 9;
    const int dy = r / 3 - 1;
    const int dx = r - (r / 3) * 3 - 1;
    const float* xp = x + (size_t)(b * CI_ + ich) * HW_;
#pragma unroll
    for (int mo = 0; mo < 4; ++mo) {
      const int m = m0 + ml0 + mo * 32;
      const int yy = (m >> 6) + dy;
      const int xx = (m & 63) + dx;
      float v = 0.f;
      if ((unsigned)yy < 64u && (unsigned)xx < 64u) v = xp[yy * 64 + xx];
      aR[ko * 4 + mo] = v;
    }
  }
}
__device__ __forceinline__ void conv_storeA(float (*As)[LDA], int t,
                                            const float aR[16]) {
  const int ml0 = t & 31, kr = t >> 5;
#pragma unroll
  for (int ko = 0; ko < 4; ++ko)
#pragma unroll
    for (int mo = 0; mo < 4; ++mo)
      As[ml0 + mo * 32][kr + ko * 8] = aR[ko * 4 + mo];
}

__global__ __launch_bounds__(256)
void conv_k(const float* __restrict__ x, const float* __restrict__ ker,
            const float* __restrict__ bias, float* __restrict__ out) {
  __shared__ float As[128][LDA];
  __shared__ float Bs[64][LDA];
  const int mtile = blockIdx.x, ntile = blockIdx.y, b = blockIdx.z;
  const int t = threadIdx.x, lane = t & 31, wid = t >> 5;
  const int wm = wid & 3, wn = wid >> 2;
  const int half = lane >> 4, lrow = lane & 15;
  const int m0 = mtile * 128, n0 = ntile * 64;

  v8f acc[2][2] = {};
  float aR[16];
  v4f bR[2];
  conv_gather(x, b, m0, 0, t, aR);
  loadB_regs(ker + (size_t)b * CO_ * KP_, KP_, n0, 0, t, bR);
  const int NCH = KP_ / KC;             // 72
  for (int kc = 0; kc < NCH; ++kc) {
    conv_storeA(As, t, aR);
    storeB_lds(Bs, t, bR);
    __syncthreads();
    if (kc + 1 < NCH) {                 // prefetch next chunk during compute
      conv_gather(x, b, m0, (kc + 1) * KC, t, aR);
      loadB_regs(ker + (size_t)b * CO_ * KP_, KP_, n0, (kc + 1) * KC, t, bR);
    }
#pragma unroll
    for (int kk = 0; kk < KC; kk += 4) {
      v2f a0 = *(const v2f*)&As[wm * 32 + lrow][kk + 2 * half];
      v2f a1 = *(const v2f*)&As[wm * 32 + 16 + lrow][kk + 2 * half];
      v2f b0 = *(const v2f*)&Bs[wn * 32 + lrow][kk + 2 * half];
      v2f b1 = *(const v2f*)&Bs[wn * 32 + 16 + lrow][kk + 2 * half];
      acc[0][0] = wmma4(a0, b0, acc[0][0]);
      acc[0][1] = wmma4(a0, b1, acc[0][1]);
      acc[1][0] = wmma4(a1, b0, acc[1][0]);
      acc[1][1] = wmma4(a1, b1, acc[1][1]);
    }
    __syncthreads();
  }
  // epilogue: silu + bias, vectorized stores along contiguous HW
#pragma unroll
  for (int mi = 0; mi < 2; ++mi)
#pragma unroll
    for (int ni = 0; ni < 2; ++ni) {
      const int o = n0 + wn * 32 + ni * 16 + lrow;
      const int m = m0 + wm * 32 + mi * 16 + half * 8;
      const float cb = bias[o];
      float* dst = out + (size_t)(b * CO_ + o) * HW_ + m;
      v8f v = acc[mi][ni];
      v4f lo, hi;
#pragma unroll
      for (int r = 0; r < 4; ++r) {
        float y0 = v[r];     lo[r] = y0 / (1.f + __expf(-y0)) + cb;
        float y1 = v[r + 4]; hi[r] = y1 / (1.f + __expf(-y1)) + cb;
      }
      *(v4f*)dst = lo;
      *(v4f*)(dst + 4) = hi;
    }
}

extern "C" void kernel_launch(void* const* d_in, const int* in_sizes, int n_in,
                              void* d_out, int out_size, void* d_ws, size_t ws_size,
                              hipStream_t stream) {
  (void)in_sizes; (void)n_in; (void)out_size; (void)ws_size;
  const float* x         = (const float*)d_in[0];
  const float* temb      = (const float*)d_in[1];
  const float* basis     = (const float*)d_in[2];
  const float* magnitude = (const float*)d_in[3];
  const float* out_lin   = (const float*)d_in[4];
  const float* affine_w  = (const float*)d_in[5];
  const float* affine_b  = (const float*)d_in[6];
  const float* conv_bias = (const float*)d_in[7];
  float* out = (float*)d_out;

  char* ws = (char*)d_ws;
  float* s   = (float*)ws;                                   // 1 KB
  float* mag = (float*)(ws + 65536);                         // 8.39 MB
  float* ker = (float*)(ws + 65536 + (size_t)B_*CI_*FD_*4);  // 37.75 MB

  affine_k<<<1, 256, 0, stream>>>(temb, affine_w, affine_b, s);
  mag_k<<<dim3(CI_, B_), 256, 0, stream>>>(magnitude, s, mag);
  gen_k<<<dim3(KP_ / 128, CO_ / 64, B_), 256, 0, stream>>>(basis, mag, out_lin, ker);
  conv_k<<<dim3(HW_ / 128, CO_ / 64, B_), 256, 0, stream>>>(x, ker, conv_bias, out);
}